// GraphWalkerMemory_16484084483471
// MI455X (gfx1250) — compile-verified
//
#include <hip/hip_runtime.h>
#include <hip/hip_bf16.h>
#include <math.h>

#define B_   8
#define N_   65536
#define K_   32
#define DS   128
#define DID  64
#define H_   4
#define L_   8
#define DQIN 64
#define FF_  512
#define NIN  4096
#define V_   32000
#define BH   32      // B*H trajectories
#define CAT  192     // DS + DID
#define QW   256     // 2*Hs*Dq
#define HPAD 260     // LDS row stride for 256-wide hidden (bank-conflict pad)
#define KPAD 132     // LDS row stride for 128-wide kk (bank-conflict pad)

// ---- workspace layout (units: 4-byte elements) ----
#define WS_INKEYS 0                              // 4096*64
#define WS_QIN    (WS_INKEYS + NIN*DQIN)         // 8*256
#define WS_VINJ   (WS_QIN + B_*H_*DQIN)          // 8*128
#define WS_MOTOR  (WS_VINJ + B_*DS)              // 8*128
#define WS_MSGS   (WS_MOTOR + B_*DS)             // 32*8*128
#define WS_PATHW  (WS_MSGS + BH*L_*DS)           // 32
#define WS_W0     (WS_PATHW + BH)                // 32
#define WS_CUR    (WS_W0 + BH)                   // 32 (int)
#define WS_SCOLS  (WS_CUR + BH)                  // 32 (int)

typedef float v2f __attribute__((ext_vector_type(2)));
typedef float v8f __attribute__((ext_vector_type(8)));

__device__ __forceinline__ v8f zero8() {
  v8f z = {0.f, 0.f, 0.f, 0.f, 0.f, 0.f, 0.f, 0.f};
  return z;
}

// D = A(16x4,f32) * B(4x16,f32) + C  -- exact f32 WMMA
__device__ __forceinline__ v8f wmma4(v2f a, v2f b, v8f c) {
  return __builtin_amdgcn_wmma_f32_16x16x4_f32(false, a, false, b, (short)0, c,
                                               false, false);
}

__device__ __forceinline__ float gelu_t(float x) {
  const float c = 0.7978845608028654f;  // sqrt(2/pi)
  float x3 = x * x * x;
  return 0.5f * x * (1.0f + tanhf(c * (x + 0.044715f * x3)));
}

// ---------------- setup: h = emb[token], q_in = h@in_q_w, v_inject = h@in_v_w
__global__ void k_embed(const int* __restrict__ token_id,
                        const float* __restrict__ token_emb,
                        const float* __restrict__ in_q_w,
                        const float* __restrict__ in_v_w, float* __restrict__ ws) {
  __shared__ float hsh[DS];
  int b = blockIdx.x, tid = threadIdx.x;
  int tk = token_id[b];
  if (tid < DS) hsh[tid] = token_emb[(size_t)tk * DS + tid];
  __syncthreads();
  {
    float acc = 0.f;
    for (int k = 0; k < DS; ++k) acc += hsh[k] * in_q_w[k * (H_ * DQIN) + tid];
    ws[WS_QIN + b * (H_ * DQIN) + tid] = acc;
  }
  if (tid < DS) {
    float acc = 0.f;
    for (int k = 0; k < DS; ++k) acc += hsh[k] * in_v_w[k * DS + tid];
    ws[WS_VINJ + b * DS + tid] = acc;
  }
}

// ---------------- in_keys = col_id[input_positions] @ in_k_w  (WMMA f32)
// grid: 4096/32 = 128 blocks, 8 waves: 2 row-tiles x 4 col-tiles
__global__ void k_inkeys(const float* __restrict__ col_id,
                         const int* __restrict__ input_positions,
                         const float* __restrict__ in_k_w, float* __restrict__ ws) {
  __shared__ int rowp[32];
  int base = blockIdx.x * 32;
  int tid = threadIdx.x;
  if (tid < 32) rowp[tid] = input_positions[base + tid];
  __syncthreads();
  int wave = tid >> 5, lane = tid & 31;
  int rowTile = wave >> 2, colTile = wave & 3;
  int lhalf = lane >> 4, l15 = lane & 15;
  int p = rowp[rowTile * 16 + l15];
  int col = colTile * 16 + l15;
  v8f acc = zero8();
  for (int kb = 0; kb < 16; ++kb) {
    int k0 = kb * 4 + lhalf * 2;
    v2f a, bf;
    a.x = col_id[(size_t)p * DID + k0];
    a.y = col_id[(size_t)p * DID + k0 + 1];
    bf.x = in_k_w[k0 * DQIN + col];
    bf.y = in_k_w[(k0 + 1) * DQIN + col];
    acc = wmma4(a, bf, acc);
  }
  for (int i = 0; i < 8; ++i) {
    int m = lhalf * 8 + i;
    ws[WS_INKEYS + (base + rowTile * 16 + m) * DQIN + col] = acc[i];
  }
}

// ---------------- input routing: per (b,h) softmax/argmax over 4096 positions
__global__ void k_route(const float* __restrict__ input_E_bias,
                        const int* __restrict__ input_positions,
                        float* __restrict__ ws) {
  __shared__ float sc[NIN];
  __shared__ float qv[DQIN];
  __shared__ float rmax[256];
  __shared__ int rarg[256];
  __shared__ float rsum[256];
  int bh = blockIdx.x, b = bh >> 2, h = bh & 3, tid = threadIdx.x;
  if (tid < DQIN) qv[tid] = ws[WS_QIN + b * (H_ * DQIN) + h * DQIN + tid];
  __syncthreads();
  for (int n = tid; n < NIN; n += 256) {
    float acc = 0.f;
    const float* kr = &ws[WS_INKEYS + n * DQIN];
    for (int k = 0; k < DQIN; ++k) acc += qv[k] * kr[k];
    sc[n] = acc * 0.125f + input_E_bias[h * NIN + n];  // 1/sqrt(64)
  }
  __syncthreads();
  float mx = -1e30f;
  int am = 0x7fffffff;
  for (int n = tid; n < NIN; n += 256) {
    float v = sc[n];
    if (v > mx || (v == mx && n < am)) { mx = v; am = n; }
  }
  rmax[tid] = mx; rarg[tid] = am;
  __syncthreads();
  for (int st = 128; st > 0; st >>= 1) {
    if (tid < st) {
      float vo = rmax[tid + st];
      if (vo > rmax[tid] || (vo == rmax[tid] && rarg[tid + st] < rarg[tid])) {
        rmax[tid] = vo; rarg[tid] = rarg[tid + st];
      }
    }
    __syncthreads();
  }
  float gmx = rmax[0];
  int gam = rarg[0];
  float se = 0.f;
  for (int n = tid; n < NIN; n += 256) se += expf(sc[n] - gmx);
  rsum[tid] = se;
  __syncthreads();
  for (int st = 128; st > 0; st >>= 1) {
    if (tid < st) rsum[tid] += rsum[tid + st];
    __syncthreads();
  }
  if (tid == 0) {
    float w0 = expf(sc[gam] - gmx) / rsum[0];  // == 1/sum at the argmax
    int scol = input_positions[gam];
    ((int*)ws)[WS_CUR + bh] = scol;
    ((int*)ws)[WS_SCOLS + bh] = scol;
    ws[WS_W0 + bh] = w0;
    ws[WS_PATHW + bh] = w0;
  }
}

// ---------------- fused 8-hop walk: one block per trajectory.
// Per hop: lazy s_eff -> rmsnorm|col_id -> MLP msg -> q head -> neighbor-key
// double GEMM (WMMA f32, staged in LDS) -> score/argmax -> advance.
__global__ void k_walk(const float* __restrict__ s,
                       const float* __restrict__ col_id,
                       const float* __restrict__ decay_w,
                       const float* __restrict__ decay_b,
                       const float* __restrict__ cn_w,
                       const float* __restrict__ mlp_w1,
                       const float* __restrict__ mlp_b1,
                       const float* __restrict__ mlp_w2,
                       const float* __restrict__ mlp_b2,
                       const float* __restrict__ q_w1,
                       const float* __restrict__ q_b1,
                       const float* __restrict__ q_w2,
                       const float* __restrict__ q_b2,
                       const float* __restrict__ k_w1,
                       const float* __restrict__ k_b1,
                       const float* __restrict__ k_w2,
                       const float* __restrict__ k_b2,
                       const int* __restrict__ out_nbrs,
                       const float* __restrict__ E_bias_flat,
                       float* __restrict__ ws) {
  __shared__ float sx[DS];
  __shared__ float cat[CAT];
  __shared__ float hid[FF_];
  __shared__ float qh[QW];
  __shared__ float qv[DS];
  __shared__ float red[256];
  __shared__ int nrow[K_];
  __shared__ float khid[K_ * HPAD];  // 32 x 260 floats
  __shared__ float kkl[K_ * KPAD];   // 32 x 132 floats
  __shared__ float scw[K_];
  __shared__ int curs;
  __shared__ float pws;

  int t = blockIdx.x, b = t >> 2, tid = threadIdx.x;
  int wave = tid >> 5, lane = tid & 31, lhalf = lane >> 4, l15 = lane & 15;
  const int* scolp = (const int*)ws + WS_SCOLS;
  if (tid == 0) {
    curs = ((const int*)ws)[WS_CUR + t];
    pws = ws[WS_PATHW + t];
  }
  __syncthreads();

  for (int hop = 0; hop < L_; ++hop) {
    int c = curs;
    // ---- lazy effective state at column c + squared-sum for rmsnorm
    float x = 0.f;
    if (tid < DS) {
      float aacc = decay_b[0];
      for (int k = 0; k < DID; ++k)
        aacc += col_id[(size_t)c * DID + k] * decay_w[k];
      float alpha = 1.f / (1.f + expf(-aacc));
      float inj = 0.f;
      for (int hh = 0; hh < H_; ++hh)
        if (scolp[b * H_ + hh] == c)
          inj += ws[WS_W0 + b * H_ + hh] * ws[WS_VINJ + b * DS + tid];
      x = s[((size_t)b * N_ + c) * DS + tid] * alpha + inj;
      sx[tid] = x;
    }
    red[tid] = (tid < DS) ? x * x : 0.f;
    __syncthreads();
    for (int st = 128; st > 0; st >>= 1) {
      if (tid < st) red[tid] += red[tid + st];
      __syncthreads();
    }
    float rn = rsqrtf(red[0] / (float)DS + 1e-6f);
    if (tid < DS) cat[tid] = sx[tid] * rn * cn_w[tid];
    else if (tid < CAT) cat[tid] = col_id[(size_t)c * DID + (tid - DS)];
    __syncthreads();
    // ---- message MLP: gelu(cat@mlp_w1+b1)@mlp_w2+b2, scaled by path weight
    for (int j = tid; j < FF_; j += 256) {
      float acc = mlp_b1[j];
      for (int k = 0; k < CAT; ++k) acc += cat[k] * mlp_w1[k * FF_ + j];
      hid[j] = gelu_t(acc);
    }
    __syncthreads();
    if (tid < DS) {
      float acc = mlp_b2[tid];
      for (int k = 0; k < FF_; ++k) acc += hid[k] * mlp_w2[k * DS + tid];
      ws[WS_MSGS + (t * L_ + hop) * DS + tid] = acc * pws;
    }
    if (hop == L_ - 1) break;
    // ---- query head: q = gelu(cat@q_w1+b1)@q_w2+b2 -> LDS
    {
      float acc = q_b1[tid];
      for (int k = 0; k < CAT; ++k) acc += cat[k] * q_w1[k * QW + tid];
      qh[tid] = gelu_t(acc);
    }
    __syncthreads();
    if (tid < DS) {
      float acc = q_b2[tid];
      for (int k = 0; k < QW; ++k) acc += qh[k] * q_w2[k * DS + tid];
      qv[tid] = acc;
    }
    if (tid < K_) nrow[tid] = out_nbrs[(size_t)c * K_ + tid];
    __syncthreads();
    // ---- GEMM1 (WMMA): (32x64)@(64x256) -> gelu -> khid (LDS)
    // 32 tiles (2 row-tiles x 16 col-tiles), 4 per wave
    for (int tt = 0; tt < 4; ++tt) {
      int tileId = wave * 4 + tt;
      int rt = tileId >> 4, ct = tileId & 15;
      int col = ct * 16 + l15;
      int p = nrow[rt * 16 + l15];
      v8f acc = zero8();
      for (int kb = 0; kb < 16; ++kb) {
        int k0 = kb * 4 + lhalf * 2;
        v2f a, bf;
        a.x = col_id[(size_t)p * DID + k0];
        a.y = col_id[(size_t)p * DID + k0 + 1];
        bf.x = k_w1[k0 * QW + col];
        bf.y = k_w1[(k0 + 1) * QW + col];
        acc = wmma4(a, bf, acc);
      }
      float bias = k_b1[col];
      for (int i = 0; i < 8; ++i) {
        int m = rt * 16 + lhalf * 8 + i;
        khid[m * HPAD + col] = gelu_t(acc[i] + bias);
      }
    }
    __syncthreads();
    // ---- GEMM2 (WMMA): (32x256)@(256x128) -> kkl (LDS)
    // 16 tiles (2 row-tiles x 8 col-tiles), 2 per wave
    for (int tt = 0; tt < 2; ++tt) {
      int tileId = wave * 2 + tt;
      int rt = tileId >> 3, ct = tileId & 7;
      int col = ct * 16 + l15;
      int arow = rt * 16 + l15;
      v8f acc = zero8();
      for (int kb = 0; kb < 64; ++kb) {
        int k0 = kb * 4 + lhalf * 2;
        v2f a, bf;
        a.x = khid[arow * HPAD + k0];
        a.y = khid[arow * HPAD + k0 + 1];
        bf.x = k_w2[k0 * DS + col];
        bf.y = k_w2[(k0 + 1) * DS + col];
        acc = wmma4(a, bf, acc);
      }
      float bias = k_b2[col];
      for (int i = 0; i < 8; ++i) {
        int m = rt * 16 + lhalf * 8 + i;
        kkl[m * KPAD + col] = acc[i] + bias;
      }
    }
    __syncthreads();
    // ---- neighbor scores + softmax/argmax, advance walk
    if (tid < K_) {
      float acc = 0.f;
      for (int d = 0; d < DS; ++d) acc += qv[d] * kkl[tid * KPAD + d];
      scw[tid] = acc * 0.17677669529663687f +  // 1/sqrt(32)
                 E_bias_flat[(size_t)c * K_ + tid];
    }
    __syncthreads();
    if (tid == 0) {
      float mx = scw[0];
      int am = 0;
      for (int j = 1; j < K_; ++j)
        if (scw[j] > mx) { mx = scw[j]; am = j; }
      float sum = 0.f;
      for (int j = 0; j < K_; ++j) sum += expf(scw[j] - mx);
      pws *= expf(scw[am] - mx) / sum;
      curs = nrow[am];
    }
    __syncthreads();
  }
}

// ---------------- motor readout: attention over 32 messages + rmsnorm
__global__ void k_motor(const float* __restrict__ out_k_w,
                        const float* __restrict__ out_v_w,
                        const float* __restrict__ motor_query,
                        const float* __restrict__ out_norm_w,
                        float* __restrict__ ws) {
  __shared__ float mr[DS];
  __shared__ float vals[BH * DS];
  __shared__ float red[DS];
  __shared__ float attw[BH];
  int b = blockIdx.x, d = threadIdx.x;  // 128 threads
  for (int n = 0; n < BH; ++n) {
    mr[d] = ws[WS_MSGS + (size_t)(b * BH + n) * DS + d];
    __syncthreads();
    float kacc = 0.f, vacc = 0.f;
    for (int k = 0; k < DS; ++k) {
      float m = mr[k];
      kacc += m * out_k_w[k * DS + d];
      vacc += m * out_v_w[k * DS + d];
    }
    vals[n * DS + d] = vacc;
    red[d] = motor_query[d] * kacc;
    __syncthreads();
    for (int st = 64; st > 0; st >>= 1) {
      if (d < st) red[d] += red[d + st];
      __syncthreads();
    }
    if (d == 0) attw[n] = red[0] * 0.08838834764831845f;  // 1/sqrt(128)
    __syncthreads();
  }
  if (d == 0) {
    float mx = attw[0];
    for (int n = 1; n < BH; ++n) mx = fmaxf(mx, attw[n]);
    float sum = 0.f;
    for (int n = 0; n < BH; ++n) { attw[n] = expf(attw[n] - mx); sum += attw[n]; }
    float inv = 1.f / sum;
    for (int n = 0; n < BH; ++n) attw[n] *= inv;
  }
  __syncthreads();
  float mo = 0.f;
  for (int n = 0; n < BH; ++n) mo += attw[n] * vals[n * DS + d];
  red[d] = mo * mo;
  __syncthreads();
  for (int st = 64; st > 0; st >>= 1) {
    if (d < st) red[d] += red[d + st];
    __syncthreads();
  }
  float rn = rsqrtf(red[0] / (float)DS + 1e-6f);
  ws[WS_MOTOR + b * DS + d] = mo * rn * out_norm_w[d];
}

// ---------------- logits = motor(8x128, padded to 16) @ token_emb^T (WMMA f32)
// grid 250 x 256: 8 waves x 16 cols = 128 cols/block
__global__ void k_logits(const float* __restrict__ token_emb,
                         const float* __restrict__ ws, float* __restrict__ out) {
  int tid = threadIdx.x;
  int wave = tid >> 5, lane = tid & 31, lhalf = lane >> 4, l15 = lane & 15;
  int colBase = (blockIdx.x * 8 + wave) * 16;
  const float* motor = &ws[WS_MOTOR];
  int arow = l15;  // rows 8..15 zero-padded
  int col = colBase + l15;
  v8f acc = zero8();
  for (int kb = 0; kb < 32; ++kb) {
    int k0 = kb * 4 + lhalf * 2;
    v2f a, bf;
    a.x = (arow < B_) ? motor[arow * DS + k0] : 0.f;
    a.y = (arow < B_) ? motor[arow * DS + k0 + 1] : 0.f;
    bf.x = token_emb[(size_t)col * DS + k0];
    bf.y = token_emb[(size_t)col * DS + k0 + 1];
    acc = wmma4(a, bf, acc);
  }
  if (lhalf == 0) {  // lanes 0..15 hold rows 0..7 = all valid batch rows
    for (int i = 0; i < 8; ++i) out[(size_t)i * V_ + col] = acc[i];
  }
}

extern "C" void kernel_launch(void* const* d_in, const int* in_sizes, int n_in,
                              void* d_out, int out_size, void* d_ws, size_t ws_size,
                              hipStream_t stream) {
  (void)in_sizes; (void)n_in; (void)out_size; (void)ws_size;
  const int*   token_id        = (const int*)d_in[0];
  const float* s               = (const float*)d_in[1];
  const float* token_emb       = (const float*)d_in[2];
  const float* col_id          = (const float*)d_in[3];
  const float* decay_w         = (const float*)d_in[4];
  const float* decay_b         = (const float*)d_in[5];
  const float* cn_w            = (const float*)d_in[6];
  const float* mlp_w1          = (const float*)d_in[7];
  const float* mlp_b1          = (const float*)d_in[8];
  const float* mlp_w2          = (const float*)d_in[9];
  const float* mlp_b2          = (const float*)d_in[10];
  const float* q_w1            = (const float*)d_in[11];
  const float* q_b1            = (const float*)d_in[12];
  const float* q_w2            = (const float*)d_in[13];
  const float* q_b2            = (const float*)d_in[14];
  const float* k_w1            = (const float*)d_in[15];
  const float* k_b1            = (const float*)d_in[16];
  const float* k_w2            = (const float*)d_in[17];
  const float* k_b2            = (const float*)d_in[18];
  const float* in_q_w          = (const float*)d_in[19];
  const float* in_k_w          = (const float*)d_in[20];
  const float* in_v_w          = (const float*)d_in[21];
  const float* input_E_bias    = (const float*)d_in[22];
  const float* E_bias_flat     = (const float*)d_in[23];
  const float* out_norm_w      = (const float*)d_in[24];
  const float* motor_query     = (const float*)d_in[25];
  const float* out_k_w         = (const float*)d_in[26];
  const float* out_v_w         = (const float*)d_in[27];
  const int*   out_nbrs        = (const int*)d_in[28];
  const int*   input_positions = (const int*)d_in[29];
  float* ws  = (float*)d_ws;
  float* out = (float*)d_out;

  k_embed<<<B_, 256, 0, stream>>>(token_id, token_emb, in_q_w, in_v_w, ws);
  k_inkeys<<<NIN / 32, 256, 0, stream>>>(col_id, input_positions, in_k_w, ws);
  k_route<<<BH, 256, 0, stream>>>(input_E_bias, input_positions, ws);
  k_walk<<<BH, 256, 0, stream>>>(s, col_id, decay_w, decay_b, cn_w,
                                 mlp_w1, mlp_b1, mlp_w2, mlp_b2,
                                 q_w1, q_b1, q_w2, q_b2,
                                 k_w1, k_b1, k_w2, k_b2,
                                 out_nbrs, E_bias_flat, ws);
  k_motor<<<B_, DS, 0, stream>>>(out_k_w, out_v_w, motor_query, out_norm_w, ws);
  k_logits<<<V_ / 128, 256, 0, stream>>>(token_emb, ws, out);
}